// fSelfRouting2d_35828617183259
// MI455X (gfx1250) — compile-verified
//
#include <hip/hip_runtime.h>
#include <hip/hip_bf16.h>
#include <math.h>

// Problem constants (fixed by the reference)
#define BATCH 32
#define INDIM 256
#define OUTDIM 256
#define NCOL 4096
#define UREP 8

typedef __attribute__((ext_vector_type(2))) float v2f;
typedef __attribute__((ext_vector_type(8))) float v8f;

// LDS layout for norms kernel: x tile, K-pair interleaved.
// Element (k, c), k in [0,256), c in [0,64):
//   float offset = (k>>1)*XS2_STRIDE + 2*c + (k&1)
// XS2_STRIDE = 160 floats (128 data + 32 pad) so the hi lane-half (kp+1) is
// shifted by 160 mod 64 = 32 banks -> conflict-free full 64-bank sweep on the
// ds_load_b64 B-operand fetches.
#define XS2_STRIDE 160
#define XS_FLOATS (128 * XS2_STRIDE)        // 20480 floats (80 KB)
#define PS_FLOATS (4 * 8 * 32)              // 1024  (col-tile, wave, lane)
#define K1_SMEM_BYTES ((XS_FLOATS + PS_FLOATS) * 4)

// ---------------------------------------------------------------------------
// Kernel 1: for each (b, n) compute a_u = ||W1 @ x[b,:,n]|| via WMMA f32 16x16x4
//           and x_norm = ||x[b,:,n]||.  One block = (batch b, 64-column tile).
//           8 waves; wave w covers OUT row-tiles {w, w+8} x 4 column-tiles.
// ---------------------------------------------------------------------------
__global__ __launch_bounds__(256) void norms_kernel(
    const float* __restrict__ x, const float* __restrict__ W1,
    float* __restrict__ s_au, float* __restrict__ s_xn) {
  extern __shared__ float smem[];
  float* xs = smem;                 // K-pair interleaved x tile
  float* ps = smem + XS_FLOATS;     // [4][8][32] squared partial sums

  const int tid  = threadIdx.x;
  const int w    = tid >> 5;        // wave 0..7
  const int lane = tid & 31;
  const int m    = lane & 15;       // M (A) / N (B,C) index inside a 16-tile
  const int hi   = lane >> 4;       // lane half selects K pair {0,1} vs {2,3}

  const int b  = blockIdx.y;
  const int n0 = blockIdx.x * 64;

  // Stage x[b, 0:256, n0:n0+64] into LDS, K-pair interleaved.
  // Thread t handles row pair p = t>>1 (rows 2p, 2p+1), columns c0..c0+31.
  {
    const int p     = tid >> 1;
    const int halfc = tid & 1;
    const int c0    = halfc * 32;
    const float4* rA4 = reinterpret_cast<const float4*>(
        x + ((size_t)(b * INDIM + 2 * p)) * NCOL + n0 + c0);
    const float4* rB4 = reinterpret_cast<const float4*>(
        x + ((size_t)(b * INDIM + 2 * p + 1)) * NCOL + n0 + c0);
    float2* xs2 = reinterpret_cast<float2*>(xs) + p * (XS2_STRIDE / 2) + c0;
#pragma unroll
    for (int q = 0; q < 8; ++q) {
      const float4 va = rA4[q];
      const float4 vb = rB4[q];
      float2 t0; t0.x = va.x; t0.y = vb.x;
      float2 t1; t1.x = va.y; t1.y = vb.y;
      float2 t2; t2.x = va.z; t2.y = vb.z;
      float2 t3; t3.x = va.w; t3.y = vb.w;
      xs2[q * 4 + 0] = t0;
      xs2[q * 4 + 1] = t1;
      xs2[q * 4 + 2] = t2;
      xs2[q * 4 + 3] = t3;
    }
  }
  __syncthreads();

  float pc0 = 0.f, pc1 = 0.f, pc2 = 0.f, pc3 = 0.f;

#pragma unroll
  for (int half = 0; half < 2; ++half) {
    const int rt = w + 8 * half;    // OUT row-tile 0..15
    v8f c0 = {}, c1 = {}, c2 = {}, c3 = {};
    const float2* wrow2 =
        reinterpret_cast<const float2*>(W1 + (size_t)(rt * 16 + m) * INDIM);
#pragma unroll 4
    for (int k4 = 0; k4 < INDIM; k4 += 4) {
      // A tile (16x4 of W1): lane holds (M=m, K=k4+2*hi, +1) as one b64 load.
      const float2 tw = wrow2[(k4 >> 1) + hi];
      v2f a; a.x = tw.x; a.y = tw.y;
      // B tiles (4x16 of x): lane reads its K-pair for column ct*16+m as one
      // ds_load_b64 each (adjacent words thanks to the interleaved layout).
      const int kp = (k4 >> 1) + hi;
      const float* bp = xs + kp * XS2_STRIDE + m * 2;
      const v2f b0 = *reinterpret_cast<const v2f*>(bp);
      const v2f b1 = *reinterpret_cast<const v2f*>(bp + 32);
      const v2f b2 = *reinterpret_cast<const v2f*>(bp + 64);
      const v2f b3 = *reinterpret_cast<const v2f*>(bp + 96);
      c0 = __builtin_amdgcn_wmma_f32_16x16x4_f32(false, a, false, b0, (short)0, c0, false, false);
      c1 = __builtin_amdgcn_wmma_f32_16x16x4_f32(false, a, false, b1, (short)0, c1, false, false);
      c2 = __builtin_amdgcn_wmma_f32_16x16x4_f32(false, a, false, b2, (short)0, c2, false, false);
      c3 = __builtin_amdgcn_wmma_f32_16x16x4_f32(false, a, false, b3, (short)0, c3, false, false);
    }
    // Square-accumulate the 16x16 u_hat tiles over the OUT (M) dimension.
    // Lane covers M values {v + 8*hi} of this row-tile, column n = ct*16+m.
#pragma unroll
    for (int v = 0; v < 8; ++v) {
      pc0 += c0[v] * c0[v];
      pc1 += c1[v] * c1[v];
      pc2 += c2[v] * c2[v];
      pc3 += c3[v] * c3[v];
    }
  }
  ps[(0 * 8 + w) * 32 + lane] = pc0;
  ps[(1 * 8 + w) * 32 + lane] = pc1;
  ps[(2 * 8 + w) * 32 + lane] = pc2;
  ps[(3 * 8 + w) * 32 + lane] = pc3;
  __syncthreads();

  // Deterministic final reduction: threads 0..63 own one column each.
  if (tid < 64) {
    const int ct = tid >> 4;
    const int n  = tid & 15;
    float au2 = 0.f;
#pragma unroll
    for (int wv = 0; wv < 8; ++wv) {
      const int base = (ct * 8 + wv) * 32;
      au2 += ps[base + n] + ps[base + n + 16];
    }
    float xn2 = 0.f;
    const float2* col2 = reinterpret_cast<const float2*>(xs) + tid;
    for (int kp = 0; kp < 128; ++kp) {
      const float2 v = col2[kp * (XS2_STRIDE / 2)];
      xn2 += v.x * v.x + v.y * v.y;
    }
    const size_t o = (size_t)b * NCOL + n0 + tid;
    s_au[o] = sqrtf(au2);
    s_xn[o] = sqrtf(xn2);
  }
}

// ---------------------------------------------------------------------------
// Kernel 2: per-batch bitonic sort of (a_u, idx) descending (stable on ties),
//           emit sorted index order, amax = sorted[0], amed = sorted[2048].
// ---------------------------------------------------------------------------
__device__ __forceinline__ bool desc_before(float va, int ia, float vb, int ib) {
  return (va > vb) || (va == vb && ia < ib);
}

__global__ __launch_bounds__(1024) void sort_kernel(
    const float* __restrict__ s_au, int* __restrict__ order,
    float* __restrict__ amax, float* __restrict__ amed) {
  __shared__ float sv[NCOL];
  __shared__ int   si[NCOL];
  const int tid = threadIdx.x;
  const int b   = blockIdx.x;
  const float* src = s_au + (size_t)b * NCOL;

  for (int i = tid; i < NCOL; i += 1024) { sv[i] = src[i]; si[i] = i; }
  __syncthreads();

  for (int k = 2; k <= NCOL; k <<= 1) {
    for (int j = k >> 1; j > 0; j >>= 1) {
      for (int t = tid; t < NCOL; t += 1024) {
        const int ixj = t ^ j;
        if (ixj > t) {
          const bool up = ((t & k) == 0);
          const float va = sv[t], vb = sv[ixj];
          const int ia = si[t], ib = si[ixj];
          const bool ab = desc_before(va, ia, vb, ib);
          if (up ? !ab : ab) {
            sv[t] = vb; sv[ixj] = va;
            si[t] = ib; si[ixj] = ia;
          }
        }
      }
      __syncthreads();
    }
  }

  for (int i = tid; i < NCOL; i += 1024) order[(size_t)b * NCOL + i] = si[i];
  if (tid == 0) {
    amax[b] = sv[0];
    amed[b] = sv[NCOL - 1 - (NCOL - 1) / 2];   // = sv[2048]
  }
}

// ---------------------------------------------------------------------------
// Kernel 3: k = floor(mean_b(amed/amax) * N)
// ---------------------------------------------------------------------------
__global__ void kprop_kernel(const float* __restrict__ amax,
                             const float* __restrict__ amed,
                             int* __restrict__ kval) {
  if (threadIdx.x == 0) {
    float s = 0.f;
    for (int b = 0; b < BATCH; ++b) s += amed[b] / amax[b];
    const float prop = s / (float)BATCH;
    *kval = (int)floorf(prop * (float)NCOL);
  }
}

// ---------------------------------------------------------------------------
// Kernel 4: per batch build dense weight vector:
//   wp[b,n] = x_norm[b,n] / (sum of x_norm over top-k)  for n in top-k, else 0
// ---------------------------------------------------------------------------
__global__ __launch_bounds__(256) void weights_kernel(
    const float* __restrict__ s_xn, const int* __restrict__ order,
    const int* __restrict__ kval, float* __restrict__ wp) {
  __shared__ float red[256];
  const int tid = threadIdx.x;
  const int b   = blockIdx.x;
  const int k   = *kval;
  const int*   ob = order + (size_t)b * NCOL;
  const float* xb = s_xn + (size_t)b * NCOL;
  float*       wb = wp + (size_t)b * NCOL;

  for (int n = tid; n < NCOL; n += 256) wb[n] = 0.f;

  float p = 0.f;
  for (int j = tid; j < k; j += 256) p += xb[ob[j]];
  red[tid] = p;
  __syncthreads();
  for (int s = 128; s > 0; s >>= 1) {
    if (tid < s) red[tid] += red[tid + s];
    __syncthreads();
  }
  const float inv = 1.f / red[0];
  for (int j = tid; j < k; j += 256) {
    const int n = ob[j];
    wb[n] = xb[n] * inv;
  }
}

// ---------------------------------------------------------------------------
// Kernel 5: xw[b,i] = sum_n wp[b,n] * x[b,i,n]   (dense, coalesced)
// ---------------------------------------------------------------------------
__global__ __launch_bounds__(256) void xweighted_kernel(
    const float* __restrict__ x, const float* __restrict__ wp,
    float* __restrict__ xw) {
  __shared__ float red[256];
  const int tid = threadIdx.x;
  const int i   = blockIdx.x;
  const int b   = blockIdx.y;
  const float* xr = x + ((size_t)(b * INDIM + i)) * NCOL;
  const float* wr = wp + (size_t)b * NCOL;
  float p = 0.f;
  for (int n = tid; n < NCOL; n += 256) p += wr[n] * xr[n];
  red[tid] = p;
  __syncthreads();
  for (int s = 128; s > 0; s >>= 1) {
    if (tid < s) red[tid] += red[tid + s];
    __syncthreads();
  }
  if (tid == 0) xw[(size_t)b * INDIM + i] = red[0];
}

// ---------------------------------------------------------------------------
// Kernel 6: pose[b,o] = sum_i W1[o,i]*xw[b,i]; out[b,u,o] = pose[b,o]
// ---------------------------------------------------------------------------
__global__ __launch_bounds__(256) void pose_kernel(
    const float* __restrict__ W1, const float* __restrict__ xw,
    float* __restrict__ out) {
  __shared__ float xv[INDIM];
  const int tid = threadIdx.x;
  const int b   = blockIdx.x;
  xv[tid] = xw[(size_t)b * INDIM + tid];
  __syncthreads();
  const float* wr = W1 + (size_t)tid * INDIM;
  float p = 0.f;
  for (int i = 0; i < INDIM; ++i) p += wr[i] * xv[i];
#pragma unroll
  for (int u = 0; u < UREP; ++u)
    out[((size_t)(b * UREP + u)) * OUTDIM + tid] = p;
}

// ---------------------------------------------------------------------------
extern "C" void kernel_launch(void* const* d_in, const int* in_sizes, int n_in,
                              void* d_out, int out_size, void* d_ws, size_t ws_size,
                              hipStream_t stream) {
  const float* x  = (const float*)d_in[0];   // (32, 256, 4096)
  const float* W1 = (const float*)d_in[1];   // (256, 256)
  // d_in[2] (W2) and d_in[3] (b2) are unused by the reference.
  float* out = (float*)d_out;                // (32, 8, 256)

  char* wsb = (char*)d_ws;
  float* s_au  = (float*)(wsb + 0);          // B*N floats   (512 KB)
  float* s_xn  = (float*)(wsb + 524288);     // B*N floats   (512 KB)
  int*   order = (int*)  (wsb + 1048576);    // B*N ints     (512 KB)
  float* wp    = (float*)(wsb + 1572864);    // B*N floats   (512 KB)
  float* xw    = (float*)(wsb + 2097152);    // B*IN floats  (32 KB)
  float* amax  = (float*)(wsb + 2129920);    // B floats
  float* amed  = (float*)(wsb + 2130048);    // B floats
  int*   kval  = (int*)  (wsb + 2130176);    // 1 int

  dim3 g1(NCOL / 64, BATCH);
  norms_kernel<<<g1, 256, K1_SMEM_BYTES, stream>>>(x, W1, s_au, s_xn);
  sort_kernel<<<BATCH, 1024, 0, stream>>>(s_au, order, amax, amed);
  kprop_kernel<<<1, 32, 0, stream>>>(amax, amed, kval);
  weights_kernel<<<BATCH, 256, 0, stream>>>(s_xn, order, kval, wp);
  xweighted_kernel<<<dim3(INDIM, BATCH), 256, 0, stream>>>(x, wp, xw);
  pose_kernel<<<BATCH, 256, 0, stream>>>(W1, xw, out);
}